// GraphSAGEBackbone_69784628625939
// MI455X (gfx1250) — compile-verified
//
#include <hip/hip_runtime.h>

typedef __attribute__((ext_vector_type(2))) float v2f;
typedef __attribute__((ext_vector_type(8))) float v8f;

#define N_NODES 50000
#define N_EDGES 800000
#define H 128
#define NLAYERS 3

// ---------------------------------------------------------------------------
// deg[dst] += 1 per edge (dst is layer-invariant: computed once)
// ---------------------------------------------------------------------------
__global__ void deg_kernel(const long long* __restrict__ ei, float* __restrict__ deg) {
    int e = blockIdx.x * blockDim.x + threadIdx.x;
    if (e < N_EDGES) {
        int dst = (int)ei[N_EDGES + e];
        atomicAdd(deg + dst, 1.0f);
    }
}

// ---------------------------------------------------------------------------
// agg[dst] += x[src]  (wave-per-edge, float4 gather, f32 atomic scatter)
// ---------------------------------------------------------------------------
__global__ void agg_kernel(const long long* __restrict__ ei,
                           const float* __restrict__ x,
                           float* __restrict__ agg) {
    int tid = blockIdx.x * blockDim.x + threadIdx.x;
    int e = tid >> 5;          // edge index (one wave-lane-group per edge)
    int q = tid & 31;          // 32 lanes * 4 floats = 128 features
    if (e >= N_EDGES) return;
    int src = (int)ei[e];
    int dst = (int)ei[N_EDGES + e];
    float4 v = *(const float4*)(x + (long long)src * H + q * 4);
    float* base = agg + (long long)dst * H + q * 4;
    atomicAdd(base + 0, v.x);
    atomicAdd(base + 1, v.y);
    atomicAdd(base + 2, v.z);
    atomicAdd(base + 3, v.w);
}

// ---------------------------------------------------------------------------
// Fused: out = (agg/deg)@Wl^T + bl + x@Wr^T ; res = x@Ws^T + bs
//        row-wise L2 normalize(out) ; atomic column sum / sumsq for BN
// One wave = one 16-row node slab, full 128-wide output.
// fp32 WMMA 16x16x4, K=128 in 32 steps; B operands staged in LDS in the
// exact per-lane float2 layout V_WMMA_F32_16X16X4_F32 wants.
// LDS layout per matrix: lds2[kb*256 + n*2 + hi] = {W[n][kb*4+hi*2], W[n][kb*4+hi*2+1]}
// ---------------------------------------------------------------------------
__global__ __launch_bounds__(256) void sage_gemm_kernel(
    const float* __restrict__ x, const float* __restrict__ agg,
    const float* __restrict__ deg,
    const float* __restrict__ Wl, const float* __restrict__ bl,
    const float* __restrict__ Wr, const float* __restrict__ Ws,
    const float* __restrict__ bs,
    float* __restrict__ outb, float* __restrict__ resb,
    float* __restrict__ colsum, float* __restrict__ colsumsq)
{
    extern __shared__ float2 lds2[];   // 3 * 8192 float2 = 192 KB
    const int tid = threadIdx.x;

    for (int i = tid; i < 3 * 8192; i += 256) {
        int m  = i >> 13;
        int r  = i & 8191;
        int kb = r >> 8;
        int j  = r & 255;
        int n  = j >> 1;
        int hi = j & 1;
        const float* Wm = (m == 0) ? Wl : (m == 1) ? Wr : Ws;
        lds2[i] = *(const float2*)(Wm + n * H + kb * 4 + hi * 2);
    }
    __syncthreads();

    const int wave = tid >> 5;
    const int lane = tid & 31;
    const int hi   = lane >> 4;     // half-wave selector (K-pairs / M+8)
    const int nloc = lane & 15;
    const int tile = blockIdx.x * 8 + wave;   // 16-row tile
    const int r0   = tile * 16;
    if (r0 >= N_NODES) return;
    const int row  = r0 + nloc;     // A-operand row owned by this lane

    const float rdeg = 1.0f / fmaxf(deg[row], 1.0f);
    const float* xrow = x   + (long long)row * H;
    const float* arow = agg + (long long)row * H;

    v8f acc_out[8];
    v8f acc_res[8];
#pragma unroll
    for (int t = 0; t < 8; ++t) { v8f z = {}; acc_out[t] = z; acc_res[t] = z; }

    for (int kb = 0; kb < 32; ++kb) {
        const int k0 = kb * 4 + hi * 2;      // A layout: lanes<16 hold K=0,1; >=16 hold K=2,3
        float2 ax2 = *(const float2*)(xrow + k0);
        float2 ag2 = *(const float2*)(arow + k0);
        v2f a_x; a_x.x = ax2.x;        a_x.y = ax2.y;
        v2f a_g; a_g.x = ag2.x * rdeg; a_g.y = ag2.y * rdeg;
        const int bidx0 = kb * 256 + nloc * 2 + hi;
#pragma unroll
        for (int t = 0; t < 8; ++t) {
            const int bi = bidx0 + t * 32;   // n = t*16 + nloc
            float2 blv = lds2[bi];
            float2 brv = lds2[8192 + bi];
            float2 bsv = lds2[16384 + bi];
            v2f b;
            b.x = blv.x; b.y = blv.y;
            acc_out[t] = __builtin_amdgcn_wmma_f32_16x16x4_f32(
                false, a_g, false, b, (short)0, acc_out[t], false, false);
            b.x = brv.x; b.y = brv.y;
            acc_out[t] = __builtin_amdgcn_wmma_f32_16x16x4_f32(
                false, a_x, false, b, (short)0, acc_out[t], false, false);
            b.x = bsv.x; b.y = bsv.y;
            acc_res[t] = __builtin_amdgcn_wmma_f32_16x16x4_f32(
                false, a_x, false, b, (short)0, acc_res[t], false, false);
        }
    }

    // --- biases (same for every accumulator row; column = t*16+nloc) ---
#pragma unroll
    for (int t = 0; t < 8; ++t) {
        float blv = bl[t * 16 + nloc];
        float bsv = bs[t * 16 + nloc];
#pragma unroll
        for (int rr = 0; rr < 8; ++rr) {
            acc_out[t][rr] += blv;
            acc_res[t][rr] += bsv;
        }
    }

    // --- row-wise L2 normalize of acc_out ---
    // C/D layout: vgpr rr, lanes<16 -> row rr, lanes>=16 -> row rr+8.
    // Reduce over columns = over 8 tiles (per lane) then over the 16-lane half.
#pragma unroll
    for (int rr = 0; rr < 8; ++rr) {
        float s = 0.0f;
#pragma unroll
        for (int t = 0; t < 8; ++t) s += acc_out[t][rr] * acc_out[t][rr];
        s += __shfl_xor(s, 1, 32);
        s += __shfl_xor(s, 2, 32);
        s += __shfl_xor(s, 4, 32);
        s += __shfl_xor(s, 8, 32);
        float scale = 1.0f / fmaxf(sqrtf(s), 1e-12f);
#pragma unroll
        for (int t = 0; t < 8; ++t) acc_out[t][rr] *= scale;
    }

    // --- store normalized out + res; accumulate BN column statistics ---
#pragma unroll
    for (int t = 0; t < 8; ++t) {
        float cs = 0.0f, cq = 0.0f;
#pragma unroll
        for (int rr = 0; rr < 8; ++rr) {
            float v = acc_out[t][rr];
            cs += v;
            cq += v * v;
            int m = rr + 8 * hi;
            long long off = (long long)(r0 + m) * H + t * 16 + nloc;
            outb[off] = v;
            resb[off] = acc_res[t][rr];
        }
        atomicAdd(colsum   + t * 16 + nloc, cs);
        atomicAdd(colsumsq + t * 16 + nloc, cq);
    }
}

// ---------------------------------------------------------------------------
// Apply BatchNorm + LeakyReLU + residual, write next-layer x
// ---------------------------------------------------------------------------
__global__ void bn_kernel(const float* __restrict__ outb,
                          const float* __restrict__ resb,
                          const float* __restrict__ colsum,
                          const float* __restrict__ colsumsq,
                          const float* __restrict__ gamma,
                          const float* __restrict__ beta,
                          float* __restrict__ xnext)
{
    long long i = (long long)blockIdx.x * blockDim.x + threadIdx.x;
    if (i >= (long long)N_NODES * H) return;
    int c = (int)(i & (H - 1));
    const float invN = 1.0f / (float)N_NODES;
    float mu  = colsum[c] * invN;
    float var = colsumsq[c] * invN - mu * mu;
    float y = (outb[i] - mu) * rsqrtf(var + 1e-5f) * gamma[c] + beta[c];
    y = (y >= 0.0f) ? y : 0.1f * y;
    xnext[i] = y + resb[i];
}

// ---------------------------------------------------------------------------
extern "C" void kernel_launch(void* const* d_in, const int* in_sizes, int n_in,
                              void* d_out, int out_size, void* d_ws, size_t ws_size,
                              hipStream_t stream) {
    const float*      x0    = (const float*)d_in[0];
    const long long*  ei    = (const long long*)d_in[1];   // int64 edge_index [2,E]
    const float*      Wl    = (const float*)d_in[2];       // [L,H,H]
    const float*      bl    = (const float*)d_in[3];       // [L,H]
    const float*      Wr    = (const float*)d_in[4];       // [L,H,H]
    const float*      gamma = (const float*)d_in[5];       // [L,H]
    const float*      beta  = (const float*)d_in[6];       // [L,H]
    const float*      Ws    = (const float*)d_in[7];       // [H,H]
    const float*      bs    = (const float*)d_in[8];       // [H]
    float* out = (float*)d_out;

    float* ws = (float*)d_ws;
    const size_t NH = (size_t)N_NODES * H;
    float* agg      = ws;
    float* outb     = agg  + NH;
    float* resb     = outb + NH;
    float* xA       = resb + NH;
    float* xB       = xA   + NH;
    float* deg      = xB   + NH;
    float* colsum   = deg  + N_NODES;
    float* colsumsq = colsum + H;

    // degree (dst only) -- identical for all layers
    hipMemsetAsync(deg, 0, N_NODES * sizeof(float), stream);
    deg_kernel<<<(N_EDGES + 255) / 256, 256, 0, stream>>>(ei, deg);

    const float* xin = x0;
    for (int l = 0; l < NLAYERS; ++l) {
        float* xout = (l == NLAYERS - 1) ? out : ((l == 0) ? xA : xB);

        hipMemsetAsync(agg, 0, NH * sizeof(float), stream);
        hipMemsetAsync(colsum, 0, H * sizeof(float), stream);
        hipMemsetAsync(colsumsq, 0, H * sizeof(float), stream);

        agg_kernel<<<(N_EDGES * 32) / 256, 256, 0, stream>>>(ei, xin, agg);

        dim3 grid((N_NODES + 127) / 128);    // 8 waves/block, 16 rows/wave
        sage_gemm_kernel<<<grid, 256, 3 * 8192 * sizeof(float2), stream>>>(
            xin, agg, deg,
            Wl + (size_t)l * H * H, bl + (size_t)l * H,
            Wr + (size_t)l * H * H, Ws, bs,
            outb, resb, colsum, colsumsq);

        bn_kernel<<<(int)((NH + 255) / 256), 256, 0, stream>>>(
            outb, resb, colsum, colsumsq,
            gamma + (size_t)l * H, beta + (size_t)l * H, xout);

        xin = xout;
    }
}